// Source_8315056685408
// MI455X (gfx1250) — compile-verified
//
#include <hip/hip_runtime.h>
#include <hip/hip_bf16.h>
#include <math.h>

typedef __attribute__((ext_vector_type(2))) float v2f;
typedef __attribute__((ext_vector_type(8))) float v8f;

#define NN    4096
#define DLAT  32
#define TILES 256      // NN / 16
#define JSPLIT 4       // column-tile splits per row strip

// ---------------------------------------------------------------------------
// Kernel 1: per-row squared norms of latent (4096 x 32)
// ---------------------------------------------------------------------------
__global__ __launch_bounds__(256) void row_sq_kernel(const float* __restrict__ latent,
                                                     float* __restrict__ sq) {
    int i = blockIdx.x * 256 + threadIdx.x;
    if (i < NN) {
        const float4* p = (const float4*)(latent + (size_t)i * DLAT);
        float s = 0.0f;
#pragma unroll
        for (int k = 0; k < DLAT / 4; ++k) {
            float4 v = p[k];
            s += v.x * v.x + v.y * v.y + v.z * v.z + v.w * v.w;
        }
        sq[i] = s;
    }
}

// ---------------------------------------------------------------------------
// Kernel 2: fused Gram (WMMA f32 16x16x4) -> distance -> masked reductions
// grid = (TILES, JSPLIT), block = 256 (8 waves). Wave w of block (ti, by)
// handles column tiles tj = by*64 + w, +8, ... (8 tiles of 16x16 each).
// Per tile: issue all streaming loads first (dis + mask), then the 8 WMMAs,
// then a lean math phase (raw v_sqrt_f32 / v_log_f32 transcendentals).
// ---------------------------------------------------------------------------
__global__ __launch_bounds__(256) void tile_loss_kernel(
    const float* __restrict__ latent,
    const float* __restrict__ dis,
    const unsigned char* __restrict__ mask,
    const float* __restrict__ sq,
    float* __restrict__ partial)
{
    const int ti   = blockIdx.x;          // row tile 0..255
    const int wave = threadIdx.x >> 5;
    const int lane = threadIdx.x & 31;
    const int hi   = lane >> 4;           // 0: lanes 0-15, 1: lanes 16-31
    const int r    = lane & 15;

    // A operand (16x4 f32 per step): lane holds A[r][4*kk + 2*hi + {0,1}]
    v2f a[8];
    {
        const float* arow = latent + (size_t)(ti * 16 + r) * DLAT + 2 * hi;
#pragma unroll
        for (int kk = 0; kk < 8; ++kk)
            a[kk] = *(const v2f*)(arow + 4 * kk);
    }
    float sqi[8];
#pragma unroll
    for (int p = 0; p < 8; ++p)
        sqi[p] = sq[ti * 16 + p + 8 * hi];   // row index of C/D element p

    const float inv_nd   = 1.0f / 28.0f;           // 1/sqrt(784)
    const float inv_nl   = 0.17677669529663689f;   // 1/sqrt(32)
    const float neg_ln2  = -0.69314718055994531f;  // e2 = -ln2 * log2(1+dls)
    const float keep_lim = 19.085536923187668f;    // e^3 - 1  (e2 > -3  <=>  dls < lim)

    float s_iso = 0.0f, s_push = 0.0f, c_knn = 0.0f;
    int   niter = 0;

    const int gi0 = ti * 16 + 8 * hi;             // base row for C/D element p
    const int tj_end = (blockIdx.y + 1) * (TILES / JSPLIT);
    for (int tj = blockIdx.y * (TILES / JSPLIT) + wave; tj < tj_end; tj += 8) {
        const int gj = tj * 16 + r;               // column index for this lane
        ++niter;

        // ---- phase 1: issue all streaming HBM loads for this tile ----
        float         dd[8];
        unsigned char mm[8];
        {
            const float*         dp = dis  + (size_t)gi0 * NN + gj;
            const unsigned char* mp = mask + (size_t)gi0 * NN + gj;
#pragma unroll
            for (int p = 0; p < 8; ++p)
                dd[p] = __builtin_nontemporal_load(dp + (size_t)p * NN);
#pragma unroll
            for (int p = 0; p < 8; ++p)
                mm[p] = __builtin_nontemporal_load(mp + (size_t)p * NN);
        }

        // ---- phase 2: Gram tile via 8 chained f32 WMMAs ----
        const float* brow = latent + (size_t)(tj * 16 + r) * DLAT + 2 * hi;
        v8f acc = {0.f, 0.f, 0.f, 0.f, 0.f, 0.f, 0.f, 0.f};
#pragma unroll
        for (int kk = 0; kk < 8; ++kk) {
            v2f b = *(const v2f*)(brow + 4 * kk);
            acc = __builtin_amdgcn_wmma_f32_16x16x4_f32(
                /*neg_a=*/false, a[kk], /*neg_b=*/false, b,
                /*c_mod=*/(short)0, acc, /*reuse_a=*/false, /*reuse_b=*/false);
        }

        // ---- phase 3: distances + masked reductions (branchless, lean) ----
        const float sqj = sq[gj];
#pragma unroll
        for (int p = 0; p < 8; ++p) {
            float d2  = sqi[p] + sqj - 2.0f * acc[p];
            d2        = fmaxf(d2, 1e-8f);
            float dls = __builtin_amdgcn_sqrtf(d2) * inv_nl;  // dis_latent/sqrt(32)

            float fm = (float)mm[p];                   // v_cvt_f32_ubyte0 (mask is 0/1)
            float nm = 1.0f - fm;

            // iso term (only kNN pairs contribute)
            float e1 = dd[p] * inv_nd - dls;
            s_iso += fm * (e1 * e1);
            c_knn += fm;

            // push term: e2 = -log1p(dls); keep if !mask && dls < e^3-1
            float e2 = neg_ln2 * __builtin_amdgcn_logf(1.0f + dls);
            float kp = (dls < keep_lim) ? nm : 0.0f;
            s_push += kp * (e2 * e2);
        }
    }
    // every element is either kNN or not: derive the non-neighbor count
    float c_not = (float)(8 * niter) - c_knn;

    // block reduction (fixed tree -> deterministic)
    __shared__ float red[4][256];
    red[0][threadIdx.x] = s_iso;
    red[1][threadIdx.x] = s_push;
    red[2][threadIdx.x] = c_knn;
    red[3][threadIdx.x] = c_not;
    __syncthreads();
    for (int off = 128; off > 0; off >>= 1) {
        if (threadIdx.x < off) {
#pragma unroll
            for (int q = 0; q < 4; ++q)
                red[q][threadIdx.x] += red[q][threadIdx.x + off];
        }
        __syncthreads();
    }
    if (threadIdx.x == 0) {
        float* out = partial + 4 * ((size_t)blockIdx.y * gridDim.x + blockIdx.x);
        out[0] = red[0][0];
        out[1] = red[1][0];
        out[2] = red[2][0];
        out[3] = red[3][0];
    }
}

// ---------------------------------------------------------------------------
// Kernel 3: final reduction over block partials + scalar loss formula
// ---------------------------------------------------------------------------
__global__ __launch_bounds__(256) void final_loss_kernel(const float* __restrict__ partial,
                                                         float* __restrict__ out,
                                                         int nparts) {
    __shared__ float red[4][256];
    float v[4] = {0.f, 0.f, 0.f, 0.f};
    for (int i = threadIdx.x; i < nparts; i += 256) {
#pragma unroll
        for (int q = 0; q < 4; ++q)
            v[q] += partial[4 * (size_t)i + q];
    }
#pragma unroll
    for (int q = 0; q < 4; ++q)
        red[q][threadIdx.x] = v[q];
    __syncthreads();
    for (int off = 128; off > 0; off >>= 1) {
        if (threadIdx.x < off) {
#pragma unroll
            for (int q = 0; q < 4; ++q)
                red[q][threadIdx.x] += red[q][threadIdx.x + off];
        }
        __syncthreads();
    }
    if (threadIdx.x == 0) {
        float S_iso  = red[0][0];
        float S_push = red[1][0];
        float Cknn   = red[2][0];
        float Cnot   = red[3][0];
        float loss_iso  = sqrtf(S_iso) / Cknn;
        float loss_push = sqrtf(S_push) / fmaxf(Cnot, 1.0f);
        // EPOCH(100) <= CHANG_START(500) -> rate = RATE_PUSH = 5.0
        out[0] = loss_iso - 5.0f * loss_push;
    }
}

// ---------------------------------------------------------------------------
extern "C" void kernel_launch(void* const* d_in, const int* in_sizes, int n_in,
                              void* d_out, int out_size, void* d_ws, size_t ws_size,
                              hipStream_t stream) {
    (void)in_sizes; (void)n_in; (void)out_size; (void)ws_size;
    // setup_inputs order: input_data [N,784] (only its shape constant matters),
    // latent_data [N,32] f32, dis_data [N,N] f32, kNN_data [N,N] bool (1 byte)
    const float*         latent = (const float*)d_in[1];
    const float*         dis    = (const float*)d_in[2];
    const unsigned char* mask   = (const unsigned char*)d_in[3];

    float* sq      = (float*)d_ws;          // 4096 floats
    float* partial = sq + NN;               // TILES*JSPLIT * 4 floats

    row_sq_kernel<<<NN / 256, 256, 0, stream>>>(latent, sq);

    dim3 grid(TILES, JSPLIT);
    tile_loss_kernel<<<grid, 256, 0, stream>>>(latent, dis, mask, sq, partial);

    final_loss_kernel<<<1, 256, 0, stream>>>(partial, (float*)d_out, TILES * JSPLIT);
}